// SpectralConv_59760174956811
// MI455X (gfx1250) — compile-verified
//
#include <hip/hip_runtime.h>

#define FEAT 128

typedef float v2f __attribute__((ext_vector_type(2)));
typedef float v8f __attribute__((ext_vector_type(8)));

// ---------------------------------------------------------------------------
// Kernel 1: out = A[M,128] @ W[128,128] using native fp32 WMMA 16x16x4.
// One wave per 16x16 output tile; block = 256 threads = 8 waves = one 16-row
// stripe covering all 8 column tiles (N=128). K-loop fully unrolled: 32 WMMAs.
// ---------------------------------------------------------------------------
__global__ __launch_bounds__(256) void gemm_xw_wmma(const float* __restrict__ A,
                                                    const float* __restrict__ W,
                                                    float* __restrict__ out,
                                                    int M) {
    const int wave = threadIdx.x >> 5;          // 0..7 -> column tile
    const int lane = threadIdx.x & 31;
    const int lm   = lane & 15;                 // row/col within tile
    const int hi   = lane >> 4;                 // 0 or 1 (K-pair select)
    const int m0   = blockIdx.x * 16;
    const int n0   = wave * 16;

    // Clamp row for safe loads if M not multiple of 16 (EXEC must stay all-1s
    // for WMMA, so we clamp instead of predicating).
    const int mrow = (m0 + lm < M) ? (m0 + lm) : (M - 1);
    const float* __restrict__ arow = A + (size_t)mrow * FEAT;

    v8f acc = {};
#pragma unroll
    for (int kt = 0; kt < 32; ++kt) {
        const int k = kt * 4;
        // A 16x4 tile: lanes 0-15 hold {K=k,k+1}, lanes 16-31 hold {K=k+2,k+3}
        v2f a = *(const v2f*)(arow + k + 2 * hi);
        // B 4x16 tile of W (row-major [K,N])
        v2f b;
        b.x = W[(size_t)(k + hi) * FEAT + n0 + lm];
        b.y = W[(size_t)(k + 2 + hi) * FEAT + n0 + lm];
        acc = __builtin_amdgcn_wmma_f32_16x16x4_f32(
            /*neg_a=*/false, a, /*neg_b=*/false, b,
            /*c_mod=*/(short)0, acc, /*reuse_a=*/false, /*reuse_b=*/false);
    }

    // C/D layout: VGPR i = row (m0 + hi*8 + i), col n0 + lm
#pragma unroll
    for (int i = 0; i < 8; ++i) {
        const int r = m0 + hi * 8 + i;
        if (r < M) out[(size_t)r * FEAT + n0 + lm] = acc[i];
    }
}

// ---------------------------------------------------------------------------
// Kernel 2: zero-fill (atomic accumulation target), float4 stores.
// ---------------------------------------------------------------------------
__global__ __launch_bounds__(256) void fill_zero(float4* __restrict__ p, size_t n4) {
    size_t i = (size_t)blockIdx.x * blockDim.x + threadIdx.x;
    const size_t stride = (size_t)gridDim.x * blockDim.x;
    for (; i < n4; i += stride) p[i] = float4{0.f, 0.f, 0.f, 0.f};
}

// ---------------------------------------------------------------------------
// Kernel 3: COO SpMM scatter: dst[rows[e],:] += vals[e] * src[cols[e],:].
// One wave per edge: lane l owns 4 consecutive floats -> one global_load_b128
// gather + 4 native global_atomic_add_f32. Both buffers are 51.2 MB and stay
// resident in the 192 MB L2, so atomics/gathers never hit HBM steady-state.
// Prefetch a future edge's source row (lowers to global_prefetch_b8).
// ---------------------------------------------------------------------------
__global__ __launch_bounds__(256) void spmm_scatter(const float* __restrict__ src,
                                                    float* __restrict__ dst,
                                                    const float* __restrict__ vals,
                                                    const int* __restrict__ rows,
                                                    const int* __restrict__ cols,
                                                    int E) {
    const int e = blockIdx.x * 8 + (threadIdx.x >> 5);   // 8 edges per block
    const int lane = threadIdx.x & 31;
    if (e >= E) return;

    const int   r = rows[e];
    const int   c = cols[e];
    const float v = vals[e];

    // Software prefetch of a future gather row into the cache hierarchy.
    const int ep = e + 4096;
    if (ep < E) {
        const int cp = cols[ep];
        __builtin_prefetch(src + (size_t)cp * FEAT + lane * 4, 0, 3);
    }

    const float4 s = ((const float4*)(src + (size_t)c * FEAT))[lane];
    float* d = dst + (size_t)r * FEAT + lane * 4;
    unsafeAtomicAdd(d + 0, s.x * v);
    unsafeAtomicAdd(d + 1, s.y * v);
    unsafeAtomicAdd(d + 2, s.z * v);
    unsafeAtomicAdd(d + 3, s.w * v);
}

// ---------------------------------------------------------------------------
// Kernel 4: out = x + buf (identity skip connection), float4 vectorized.
// ---------------------------------------------------------------------------
__global__ __launch_bounds__(256) void add_identity(const float4* __restrict__ x,
                                                    const float4* __restrict__ buf,
                                                    float4* __restrict__ out, size_t n4) {
    size_t i = (size_t)blockIdx.x * blockDim.x + threadIdx.x;
    const size_t stride = (size_t)gridDim.x * blockDim.x;
    for (; i < n4; i += stride) {
        const float4 a = x[i];
        const float4 b = buf[i];
        out[i] = float4{a.x + b.x, a.y + b.y, a.z + b.z, a.w + b.w};
    }
}

extern "C" void kernel_launch(void* const* d_in, const int* in_sizes, int n_in,
                              void* d_out, int out_size, void* d_ws, size_t ws_size,
                              hipStream_t stream) {
    const float* x    = (const float*)d_in[0];   // [N,128]
    const float* w    = (const float*)d_in[1];   // [128,128]
    const float* vals = (const float*)d_in[2];   // [E]
    const int*   rows = (const int*)d_in[3];     // [E]
    const int*   cols = (const int*)d_in[4];     // [E]

    const int NC = in_sizes[0];
    const int N  = NC / FEAT;
    const int E  = in_sizes[2];

    float* out = (float*)d_out;
    float* ws  = (float*)d_ws;   // one [N,128] fp32 buffer (51.2 MB)

    // 1) ws = x @ w  (fp32 WMMA)
    gemm_xw_wmma<<<(N + 15) / 16, 256, 0, stream>>>(x, w, ws, N);

    // 2) 8 propagation steps, ping-pong ws <-> out (both L2-resident)
    const int blocksE = (E + 7) / 8;
    for (int s = 0; s < 8; ++s) {
        float* srcb = (s & 1) ? out : ws;
        float* dstb = (s & 1) ? ws : out;
        fill_zero<<<2048, 256, 0, stream>>>((float4*)dstb, (size_t)NC / 4);
        spmm_scatter<<<blocksE, 256, 0, stream>>>(srcb, dstb, vals, rows, cols, E);
    }

    // 3) after 8 (even) steps the propagated signal is in ws: out = x + ws
    add_identity<<<2048, 256, 0, stream>>>((const float4*)x, (const float4*)ws,
                                           (float4*)out, (size_t)NC / 4);
}